// DEQFixedPoint_61065845014914
// MI455X (gfx1250) — compile-verified
//
#include <hip/hip_runtime.h>
#include <cmath>

// ---------------------------------------------------------------------------
// Anderson acceleration DEQ fixed point on gfx1250 (MI455X).
// f(z) = tanh(conv3x3(z) + phi + lap), B=4, C=32, H=W=128, fp32.
// Conv implemented as implicit GEMM with V_WMMA_F32_16X16X4_F32 (exact fp32).
// Boundary handling: clamped addresses + multiplicative 0/1 mask so all loads
// are unconditional (no EXEC-branch regions; WMMA requires EXEC all-ones).
// kh kept as a real loop (body = kw x cs = 48 WMMAs) so register demand fits
// in the 256-VGPR direct window: no scratch spills, no s_set_vgpr_msb.
// ---------------------------------------------------------------------------

#define B_      4
#define C_      32
#define H_      128
#define W_      128
#define HW_     (H_ * W_)          // 16384
#define FLAT_   (C_ * HW_)         // 524288
#define FLAT4_  (FLAT_ / 4)        // 131072 float4
#define M_      5                  // Anderson memory
#define MAXIT_  18
#define RIDGE_  1e-4f
#define NCHUNK_ 256                // gram stage-1 chunks per batch

typedef float v2f __attribute__((ext_vector_type(2)));
typedef float v8f __attribute__((ext_vector_type(8)));

// ---------------------------------------------------------------------------
// conv3x3 + phi + lap + tanh, one wave computes a 32x16 (co x pixel) tile as two 16x16 D tiles.
// Fragment layouts per CDNA5 ISA 7.12.2 (wave32):
//   A 16x4  : lane&15 = M row; lane>=16 -> K pair {2,3} else {0,1}; v0=Klo,v1=Khi
//   B 4x16  : lane&15 = N col; lane>=16 -> K pair {2,3}; v0=Klo row, v1=Khi row
//   C/D 16x16 f32: VGPR r -> M=r (lanes 0-15) / M=r+8 (lanes 16-31), N=lane&15
// ---------------------------------------------------------------------------
__global__ __launch_bounds__(128, 4)
void conv_tanh_wmma(const float* __restrict__ src, long srcBatchStride, long srcOff,
                    float* __restrict__ dst, long dstBatchStride, long dstOff,
                    const float* __restrict__ phi, const float* __restrict__ lap,
                    const float* __restrict__ Wc)
{
    const int lane = threadIdx.x & 31;
    const int wv   = threadIdx.x >> 5;
    const int tile = blockIdx.x * 4 + wv;      // 0 .. 4095
    const int wt   = tile & 7;                 // W/16 tiles
    const int h    = (tile >> 3) & 127;
    const int b    = tile >> 10;
    const int w0   = wt * 16;
    const int nn   = lane & 15;                // N column / A row (M)
    const int duo  = lane >> 4;                // selects K pair {0,1} vs {2,3}

    const float* zb = src + (size_t)b * srcBatchStride + srcOff;

    v8f acc0 = {};   // output channels 0..15
    v8f acc1 = {};   // output channels 16..31

    #pragma unroll 1
    for (int kh = 0; kh < 3; ++kh) {
        const int  hh  = h + kh - 1;
        const bool hok = (unsigned)hh < (unsigned)H_;
        const int  hcl = hh < 0 ? 0 : (hh > H_ - 1 ? H_ - 1 : hh);
        #pragma unroll
        for (int kw = 0; kw < 3; ++kw) {
            const int  ww  = w0 + nn + kw - 1;
            const bool ok  = hok && ((unsigned)ww < (unsigned)W_);
            const int  wcl = ww < 0 ? 0 : (ww > W_ - 1 ? W_ - 1 : ww);
            const float msk = ok ? 1.0f : 0.0f;       // unconditional loads + mask
            const float* zrow = zb + (size_t)hcl * W_ + wcl;   // + ci*HW_
            const float* wtap = Wc + kh * 3 + kw;              // + (co*32+ci)*9
            #pragma unroll
            for (int cs = 0; cs < 8; ++cs) {
                const int kc = cs * 4 + duo * 2;               // input channel base
                v2f bf;
                bf.x = msk * zrow[(size_t)kc * HW_];
                bf.y = msk * zrow[(size_t)(kc + 1) * HW_];
                const float* wp0 = wtap + ((size_t)nn * 32 + kc) * 9;
                v2f af0; af0.x = wp0[0]; af0.y = wp0[9];
                const float* wp1 = wtap + ((size_t)(nn + 16) * 32 + kc) * 9;
                v2f af1; af1.x = wp1[0]; af1.y = wp1[9];
                acc0 = __builtin_amdgcn_wmma_f32_16x16x4_f32(
                           false, af0, false, bf, (short)0, acc0, false, false);
                acc1 = __builtin_amdgcn_wmma_f32_16x16x4_f32(
                           false, af1, false, bf, (short)0, acc1, false, false);
            }
        }
    }

    // Epilogue: D[r] holds row M=r (lanes 0-15) / M=r+8 (lanes 16-31).
    float* db = dst + (size_t)b * dstBatchStride + dstOff;
    const size_t pbase = (size_t)b * C_ * HW_;
    const int col = w0 + nn;
    #pragma unroll
    for (int r = 0; r < 8; ++r) {
        const int row = r + duo * 8;
        {
            const size_t off = (size_t)row * HW_ + (size_t)h * W_ + col;
            db[off] = tanhf(acc0[r] + phi[pbase + off] + lap[pbase + off]);
        }
        {
            const size_t off = (size_t)(row + 16) * HW_ + (size_t)h * W_ + col;
            db[off] = tanhf(acc1[r] + phi[pbase + off] + lap[pbase + off]);
        }
    }
}

// ---------------------------------------------------------------------------
// Strided batch copy (float4) into/out of history tensors. Strides/offsets in
// float4 units.
// ---------------------------------------------------------------------------
__global__ __launch_bounds__(256)
void copy_hist4(const float4* __restrict__ src, long sBS, long sOff,
                float4* __restrict__ dst, long dBS, long dOff)
{
    size_t i = (size_t)blockIdx.x * blockDim.x + threadIdx.x;   // 0..B*FLAT4
    const int b = (int)(i / FLAT4_);
    const int f = (int)(i % FLAT4_);
    dst[(size_t)b * dBS + dOff + f] = src[(size_t)b * sBS + sOff + f];
}

// ---------------------------------------------------------------------------
// Gram stage 1: per (batch, chunk) partial sums of G_i . G_j, G = Fh - X.
// float4 loads; deterministic (no float atomics). Pair idx: i*(i+1)/2 + j.
// ---------------------------------------------------------------------------
__global__ __launch_bounds__(256)
void gram_stage1(const float4* __restrict__ X, const float4* __restrict__ F,
                 float* __restrict__ partial, int n)
{
    const int b     = blockIdx.y;
    const int chunk = blockIdx.x;
    const int tid   = threadIdx.x;
    const int npairs = n * (n + 1) / 2;

    float p[15];
    #pragma unroll
    for (int i = 0; i < 15; ++i) p[i] = 0.0f;

    const size_t base = (size_t)b * M_ * FLAT4_;
    const int f0 = chunk * (FLAT4_ / NCHUNK_);          // 512 float4 per chunk
    for (int t = 0; t < (FLAT4_ / NCHUNK_) / 256; ++t) {  // 2 iterations
        const int f = f0 + t * 256 + tid;
        float4 g[5];
        #pragma unroll
        for (int i = 0; i < 5; ++i) {
            if (i < n) {
                const float4 fv = F[base + (size_t)i * FLAT4_ + f];
                const float4 xv = X[base + (size_t)i * FLAT4_ + f];
                g[i].x = fv.x - xv.x; g[i].y = fv.y - xv.y;
                g[i].z = fv.z - xv.z; g[i].w = fv.w - xv.w;
            } else {
                g[i].x = g[i].y = g[i].z = g[i].w = 0.0f;
            }
        }
        int idx = 0;
        #pragma unroll
        for (int i = 0; i < 5; ++i)
            #pragma unroll
            for (int j = 0; j <= i; ++j) {
                if (i < n)
                    p[idx] += g[i].x * g[j].x + g[i].y * g[j].y +
                              g[i].z * g[j].z + g[i].w * g[j].w;
                ++idx;
            }
    }

    __shared__ float red[256];
    for (int pi = 0; pi < npairs; ++pi) {
        red[tid] = p[pi];
        __syncthreads();
        for (int s = 128; s > 0; s >>= 1) {
            if (tid < s) red[tid] += red[tid + s];
            __syncthreads();
        }
        if (tid == 0) partial[((size_t)b * NCHUNK_ + chunk) * 16 + pi] = red[0];
        __syncthreads();
    }
}

// Gram stage 2: fixed-order serial sum over chunks -> gram[b*16 + pair].
__global__ void gram_stage2(const float* __restrict__ partial,
                            float* __restrict__ gram, int n)
{
    const int tid = threadIdx.x;
    const int npairs = n * (n + 1) / 2;
    const int total = B_ * npairs;
    for (int o = tid; o < total; o += blockDim.x) {
        const int b = o / npairs, pi = o % npairs;
        float s = 0.0f;
        for (int c = 0; c < NCHUNK_; ++c)
            s += partial[((size_t)b * NCHUNK_ + c) * 16 + pi];
        gram[b * 16 + pi] = s;
    }
}

// ---------------------------------------------------------------------------
// Solve H * v = e0, H = [[0, 1s],[1s, gram + ridge*I]], alpha = v[1:].
// Gauss-Jordan with partial pivoting (H[0][0]==0 -> pivoting required).
// One thread per batch.
// ---------------------------------------------------------------------------
__global__ void solve_kernel(const float* __restrict__ gram,
                             float* __restrict__ alpha, int n)
{
    const int b = threadIdx.x;
    if (b >= B_) return;
    const int dim = n + 1;

    float A[6][7];
    for (int i = 0; i < 6; ++i)
        for (int j = 0; j < 7; ++j) A[i][j] = 0.0f;
    for (int j = 1; j < dim; ++j) { A[0][j] = 1.0f; A[j][0] = 1.0f; }
    for (int i = 1; i < dim; ++i)
        for (int j = 1; j < dim; ++j) {
            const int ii = i - 1, jj = j - 1;
            const int lo = (ii < jj) ? ii : jj;
            const int hi = (ii < jj) ? jj : ii;
            float v = gram[b * 16 + hi * (hi + 1) / 2 + lo];
            if (ii == jj) v += RIDGE_;
            A[i][j] = v;
        }
    A[0][dim] = 1.0f;   // rhs = e0

    for (int c = 0; c < dim; ++c) {
        int piv = c; float mv = fabsf(A[c][c]);
        for (int r = c + 1; r < dim; ++r) {
            const float av = fabsf(A[r][c]);
            if (av > mv) { mv = av; piv = r; }
        }
        if (piv != c)
            for (int j = 0; j <= dim; ++j) {
                const float t = A[c][j]; A[c][j] = A[piv][j]; A[piv][j] = t;
            }
        const float inv = 1.0f / A[c][c];
        for (int j = 0; j <= dim; ++j) A[c][j] *= inv;
        for (int r = 0; r < dim; ++r) {
            if (r == c) continue;
            const float fct = A[r][c];
            for (int j = 0; j <= dim; ++j) A[r][j] -= fct * A[c][j];
        }
    }
    for (int i = 0; i < n; ++i) alpha[b * 8 + i] = A[i + 1][dim];
}

// ---------------------------------------------------------------------------
// mix = sum_i alpha[i] * Fh[:, i]   (beta == 1.0). float4 path. Writes
// X[:, slot]; also writes d_out on the final iteration (output = last mix).
// ---------------------------------------------------------------------------
__global__ __launch_bounds__(256)
void mix_kernel4(const float4* __restrict__ F, const float* __restrict__ alpha,
                 float4* __restrict__ X, int slot, int n, float4* __restrict__ out)
{
    size_t i = (size_t)blockIdx.x * blockDim.x + threadIdx.x;
    const int b = (int)(i / FLAT4_);
    const int f = (int)(i % FLAT4_);
    const size_t base = (size_t)b * M_ * FLAT4_;
    float4 s; s.x = s.y = s.z = s.w = 0.0f;
    #pragma unroll
    for (int k = 0; k < 5; ++k) {
        if (k < n) {
            const float a = alpha[b * 8 + k];
            const float4 fv = F[base + (size_t)k * FLAT4_ + f];
            s.x += a * fv.x; s.y += a * fv.y; s.z += a * fv.z; s.w += a * fv.w;
        }
    }
    X[base + (size_t)slot * FLAT4_ + f] = s;
    if (out) out[(size_t)b * FLAT4_ + f] = s;
}

// ---------------------------------------------------------------------------
extern "C" void kernel_launch(void* const* d_in, const int* in_sizes, int n_in,
                              void* d_out, int out_size, void* d_ws, size_t ws_size,
                              hipStream_t stream)
{
    const float* phi = (const float*)d_in[0];
    const float* lap = (const float*)d_in[1];
    const float* x0  = (const float*)d_in[2];
    const float* Wc  = (const float*)d_in[3];
    float* out = (float*)d_out;

    float* X       = (float*)d_ws;                          // B*M*FLAT
    float* Fh      = X  + (size_t)B_ * M_ * FLAT_;          // B*M*FLAT
    float* partial = Fh + (size_t)B_ * M_ * FLAT_;          // B*NCHUNK*16
    float* gram    = partial + (size_t)B_ * NCHUNK_ * 16;   // B*16
    float* alpha   = gram + B_ * 16;                        // B*8

    const long HB  = (long)M_ * FLAT_;    // history batch stride (elements)
    const long HB4 = (long)M_ * FLAT4_;   // history batch stride (float4)

    const dim3 cblk(128), cgrd(1024);                  // 4096 wave-tiles
    const dim3 vblk(256), vgrd(B_ * FLAT4_ / 256);     // float4 elementwise

    // X[:,0] = x0 ; Fh[:,0] = f(x0)
    copy_hist4<<<vgrd, vblk, 0, stream>>>((const float4*)x0, FLAT4_, 0,
                                          (float4*)X, HB4, 0);
    conv_tanh_wmma<<<cgrd, cblk, 0, stream>>>(x0, FLAT_, 0, Fh, HB, 0, phi, lap, Wc);
    // X[:,1] = Fh[:,0] ; Fh[:,1] = f(X[:,1])
    copy_hist4<<<vgrd, vblk, 0, stream>>>((const float4*)Fh, HB4, 0,
                                          (float4*)X, HB4, FLAT4_);
    conv_tanh_wmma<<<cgrd, cblk, 0, stream>>>(X, HB, FLAT_, Fh, HB, FLAT_,
                                              phi, lap, Wc);

    for (int k = 2; k < MAXIT_; ++k) {
        const int n = (k < M_) ? k : M_;
        dim3 g1(NCHUNK_, B_);
        gram_stage1<<<g1, 256, 0, stream>>>((const float4*)X, (const float4*)Fh,
                                            partial, n);
        gram_stage2<<<1, 64, 0, stream>>>(partial, gram, n);
        solve_kernel<<<1, 32, 0, stream>>>(gram, alpha, n);
        const int slot = k % M_;
        mix_kernel4<<<vgrd, vblk, 0, stream>>>((const float4*)Fh, alpha,
                                               (float4*)X, slot, n,
                                               (k == MAXIT_ - 1) ? (float4*)out
                                                                 : nullptr);
        if (k < MAXIT_ - 1)   // output is the final mix; last f-eval is dead code
            conv_tanh_wmma<<<cgrd, cblk, 0, stream>>>(X, HB, (long)slot * FLAT_,
                                                      Fh, HB, (long)slot * FLAT_,
                                                      phi, lap, Wc);
    }
}